// AGCRNCell_63565515981251
// MI455X (gfx1250) — compile-verified
//
#include <hip/hip_runtime.h>
#include <math.h>

// AGCRN cell on MI455X (gfx1250), fp32 via V_WMMA_F32_16X16X4_F32.
// Needs ws_size >= 218,103,808 bytes (~208 MiB).
// LDS layouts are chosen so every WMMA operand is a contiguous, 8B-aligned
// v2f -> single ds_load_b64 per operand, no register marshaling.

typedef float v2f __attribute__((ext_vector_type(2)));
typedef float v8f __attribute__((ext_vector_type(8)));

#define NN   2048   // vertices
#define BB   64     // batch
#define DD   16     // embed dim
#define CIN  96     // 32 + 64
#define KTOT 288    // cheb_k * CIN
#define GOUT 128    // gate out cols
#define UOUT 64     // update out cols
#define HID  64

static __device__ __forceinline__ v8f wmma4(v2f a, v2f b, v8f c) {
  // D = A(16x4) * B(4x16) + C, fp32.
  return __builtin_amdgcn_wmma_f32_16x16x4_f32(false, a, false, b, (short)0, c,
                                               false, false);
}
static __device__ __forceinline__ v2f ld2s(const float* p) {  // LDS/global v2f
  return *(const v2f*)p;
}

// ---------------- cat = concat(x, state) : [B][N][96] ----------------
__global__ __launch_bounds__(256)
void k_build_cat(const float* __restrict__ x, const float* __restrict__ st,
                 float* __restrict__ cat) {
  int idx = blockIdx.x * 256 + threadIdx.x;          // B*N*96 total, exact
  int c  = idx % CIN;
  int bn = idx / CIN;
  cat[idx] = (c < 32) ? x[bn * 32 + c] : st[bn * HID + (c - 32)];
}

// ------------- S1 = softmax(relu(ne @ ne^T)) row-wise ----------------
__global__ __launch_bounds__(256)
void k_supports(const float* __restrict__ ne, float* __restrict__ S1) {
  __shared__ float red[256];
  int row = blockIdx.x, t = threadIdx.x;
  float er[DD];
#pragma unroll
  for (int d = 0; d < DD; ++d) er[d] = ne[row * DD + d];
  float vals[8];
  float mx = -1e30f;
#pragma unroll
  for (int i = 0; i < 8; ++i) {
    int m = t + 256 * i;
    const float* em = ne + m * DD;
    float dot = 0.f;
#pragma unroll
    for (int d = 0; d < DD; ++d) dot += er[d] * em[d];
    dot = dot > 0.f ? dot : 0.f;                     // relu
    vals[i] = dot;
    mx = fmaxf(mx, dot);
  }
  red[t] = mx; __syncthreads();
  for (int s = 128; s > 0; s >>= 1) { if (t < s) red[t] = fmaxf(red[t], red[t + s]); __syncthreads(); }
  float rmax = red[0]; __syncthreads();
  float sum = 0.f;
#pragma unroll
  for (int i = 0; i < 8; ++i) { vals[i] = __expf(vals[i] - rmax); sum += vals[i]; }
  red[t] = sum; __syncthreads();
  for (int s = 128; s > 0; s >>= 1) { if (t < s) red[t] += red[t + s]; __syncthreads(); }
  float inv = 1.f / red[0];
#pragma unroll
  for (int i = 0; i < 8; ++i) S1[row * NN + t + 256 * i] = vals[i] * inv;
}

// ---------------- S2 = 2 * S1@S1 - I  (2048^3 fp32 GEMM) ----------------
// Block: 128 rows x 64 cols. Wave: 2 row-tiles x 2 col-tiles.
__global__ __launch_bounds__(256)
void k_cheb2(const float* __restrict__ P, float* __restrict__ S2) {
  __shared__ float As[128][34];   // [row][k], even pitch -> 8B-aligned pairs
  __shared__ float Bs[64][34];    // [col][k]
  int t = threadIdx.x;
  int n0 = blockIdx.x * 128, c0 = blockIdx.y * 64;
  int w = t >> 5, lane = t & 31, lm = lane & 15, kh = lane >> 4;
  int rp = w >> 1, h = w & 1;                       // row-tile pair, col half
  v8f z = {0.f,0.f,0.f,0.f,0.f,0.f,0.f,0.f};
  v8f acc[2][2] = {{z, z}, {z, z}};
  for (int m0 = 0; m0 < NN; m0 += 32) {
    __syncthreads();
    for (int e = t; e < 128 * 32; e += 256) { int rr = e >> 5, k = e & 31; As[rr][k] = P[(n0 + rr) * NN + m0 + k]; }
    for (int e = t; e < 32 * 64;  e += 256) { int k = e >> 6, c = e & 63;  Bs[c][k]  = P[(m0 + k)  * NN + c0 + c]; }
    __syncthreads();
    for (int kk = 0; kk < 32; kk += 4) {
      v2f a0 = ld2s(&As[16 * (2 * rp + 0) + lm][kk + 2 * kh]);
      v2f a1 = ld2s(&As[16 * (2 * rp + 1) + lm][kk + 2 * kh]);
#pragma unroll
      for (int jj = 0; jj < 2; ++jj) {
        v2f b = ld2s(&Bs[16 * (2 * h + jj) + lm][kk + 2 * kh]);
        acc[0][jj] = wmma4(a0, b, acc[0][jj]);
        acc[1][jj] = wmma4(a1, b, acc[1][jj]);
      }
    }
  }
#pragma unroll
  for (int i = 0; i < 2; ++i) {
#pragma unroll
    for (int jj = 0; jj < 2; ++jj) {
#pragma unroll
      for (int v = 0; v < 8; ++v) {
        int gr = n0 + 16 * (2 * rp + i) + v + 8 * kh;
        int gc = c0 + 16 * (2 * h + jj) + lm;
        S2[gr * NN + gc] = 2.f * acc[i][jj][v] - (gr == gc ? 1.f : 0.f);
      }
    }
  }
}

// ---- xg[n][b][96] = S(2048x2048) @ cat[b](2048x96), per batch b ----
// Block: 128 rows x 96 cols. Wave: 2 row-tiles x 3 col-tiles.
__global__ __launch_bounds__(256)
void k_prop(const float* __restrict__ S, const float* __restrict__ cat,
            float* __restrict__ xg) {
  __shared__ float As[128][34];   // [row][k]
  __shared__ float Bs[96][34];    // [col][k]
  int t = threadIdx.x;
  int n0 = blockIdx.x * 128, b = blockIdx.y;
  int w = t >> 5, lane = t & 31, lm = lane & 15, kh = lane >> 4;
  int rp = w >> 1, h = w & 1;                       // row-tile pair, 3-col half
  v8f z = {0.f,0.f,0.f,0.f,0.f,0.f,0.f,0.f};
  v8f acc[2][3] = {{z, z, z}, {z, z, z}};
  const float* catb = cat + (size_t)b * NN * CIN;
  for (int m0 = 0; m0 < NN; m0 += 32) {
    __syncthreads();
    for (int e = t; e < 128 * 32; e += 256) { int rr = e >> 5, k = e & 31; As[rr][k] = S[(n0 + rr) * NN + m0 + k]; }
    for (int e = t; e < 32 * 96;  e += 256) { int k = e / 96, c = e % 96;  Bs[c][k]  = catb[(m0 + k) * CIN + c]; }
    __syncthreads();
    for (int kk = 0; kk < 32; kk += 4) {
      v2f a0 = ld2s(&As[16 * (2 * rp + 0) + lm][kk + 2 * kh]);
      v2f a1 = ld2s(&As[16 * (2 * rp + 1) + lm][kk + 2 * kh]);
#pragma unroll
      for (int jj = 0; jj < 3; ++jj) {
        v2f b2 = ld2s(&Bs[16 * (3 * h + jj) + lm][kk + 2 * kh]);
        acc[0][jj] = wmma4(a0, b2, acc[0][jj]);
        acc[1][jj] = wmma4(a1, b2, acc[1][jj]);
      }
    }
  }
#pragma unroll
  for (int i = 0; i < 2; ++i) {
#pragma unroll
    for (int jj = 0; jj < 3; ++jj) {
      int cb = 16 * (3 * h + jj);
#pragma unroll
      for (int v = 0; v < 8; ++v) {
        int gn = n0 + 16 * (2 * rp + i) + v + 8 * kh;
        xg[(gn * BB + b) * CIN + cb + lm] = acc[i][jj][v];  // [n][b][96]
      }
    }
  }
}

// ---- gate: z_r = sigmoid(Xg_n @ W_n + b_n), W_n generated from pool ----
// Xg_n columns: [0,96)=cat(k=0 identity), [96,192)=xg1, [192,288)=xg2.
// Epilogue: cat[:, n, 32:96] <- z * state (candidate in place); r -> rbuf.
__global__ __launch_bounds__(256)
void k_gate(const float* __restrict__ ne, const float* __restrict__ gw,
            const float* __restrict__ gb, float* cat,
            const float* __restrict__ xg1, const float* __restrict__ xg2,
            const float* __restrict__ state, float* __restrict__ rbuf) {
  __shared__ float nes[DD];
  __shared__ float Wp[GOUT][18];                    // [col][k], panel of 16 k
  __shared__ float bias_s[GOUT];
  int n = blockIdx.x, t = threadIdx.x;
  if (t < DD) nes[t] = ne[n * DD + t];
  __syncthreads();
  if (t < GOUT) {
    float s = 0.f;
#pragma unroll
    for (int d = 0; d < DD; ++d) s += nes[d] * gb[d * GOUT + t];
    bias_s[t] = s;
  }
  int w = t >> 5, lane = t & 31, lm = lane & 15, kh = lane >> 4;
  int cb = 16 * w;                                  // wave = col tile (8 of 16)
  v8f z8 = {0.f,0.f,0.f,0.f,0.f,0.f,0.f,0.f};
  v8f acc[4] = {z8, z8, z8, z8};
  for (int kc = 0; kc < 3; ++kc) {
    const float* abase; int astride;
    if (kc == 0) { abase = cat + n * CIN;                  astride = NN * CIN; }
    else         { abase = (kc == 1 ? xg1 : xg2) + n * BB * CIN; astride = CIN; }
    const float* pbase = gw + kc * CIN * GOUT;            // [d][288][128] slab
    for (int kk0 = 0; kk0 < CIN; kk0 += 16) {
      __syncthreads();
      for (int e = t; e < 16 * GOUT; e += 256) {          // gen W panel 16x128
        int tl = e >> 7, o = e & 127;
        float s = 0.f;
#pragma unroll
        for (int d = 0; d < DD; ++d) s += nes[d] * pbase[d * KTOT * GOUT + (kk0 + tl) * GOUT + o];
        Wp[o][tl] = s;                                    // transposed store
      }
      __syncthreads();
      for (int kk = 0; kk < 16; kk += 4) {
        int kl = kk0 + kk;
        v2f bop = ld2s(&Wp[cb + lm][kk + 2 * kh]);
#pragma unroll
        for (int r = 0; r < 4; ++r) {
          int bRow = 16 * r + lm;
          v2f aop = ld2s(&abase[bRow * astride + kl + 2 * kh]);
          acc[r] = wmma4(aop, bop, acc[r]);
        }
      }
    }
  }
#pragma unroll
  for (int r = 0; r < 4; ++r) {
#pragma unroll
    for (int v = 0; v < 8; ++v) {
      int bRow = 16 * r + v + 8 * kh;
      int col  = cb + lm;
      float s = 1.f / (1.f + __expf(-(acc[r][v] + bias_s[col])));
      int bn = bRow * NN + n;
      if (col < HID) cat[bn * CIN + 32 + col] = s * state[bn * HID + col]; // z*state
      else           rbuf[bn * HID + (col - HID)] = s;                     // r
    }
  }
}

// ---- update: hc = tanh(Xg_n @ Wu_n + bu_n); out = r*state + (1-r)*hc ----
__global__ __launch_bounds__(256)
void k_upd(const float* __restrict__ ne, const float* __restrict__ uw,
           const float* __restrict__ ub, const float* __restrict__ cat,
           const float* __restrict__ xg1, const float* __restrict__ xg2,
           const float* __restrict__ state, const float* __restrict__ rbuf,
           float* __restrict__ out) {
  __shared__ float nes[DD];
  __shared__ float Wp[UOUT][18];                    // [col][k]
  __shared__ float bias_s[UOUT];
  int n = blockIdx.x, t = threadIdx.x;
  if (t < DD) nes[t] = ne[n * DD + t];
  __syncthreads();
  if (t < UOUT) {
    float s = 0.f;
#pragma unroll
    for (int d = 0; d < DD; ++d) s += nes[d] * ub[d * UOUT + t];
    bias_s[t] = s;
  }
  int w = t >> 5, lane = t & 31, lm = lane & 15, kh = lane >> 4;
  int cb = 16 * (w & 3);                            // 4 col tiles
  int rbase = 2 * (w >> 2);                         // 2 row tiles per wave
  v8f z8 = {0.f,0.f,0.f,0.f,0.f,0.f,0.f,0.f};
  v8f acc[2] = {z8, z8};
  for (int kc = 0; kc < 3; ++kc) {
    const float* abase; int astride;
    if (kc == 0) { abase = cat + n * CIN;                  astride = NN * CIN; }
    else         { abase = (kc == 1 ? xg1 : xg2) + n * BB * CIN; astride = CIN; }
    const float* pbase = uw + kc * CIN * UOUT;            // [d][288][64] slab
    for (int kk0 = 0; kk0 < CIN; kk0 += 16) {
      __syncthreads();
      for (int e = t; e < 16 * UOUT; e += 256) {
        int tl = e >> 6, o = e & 63;
        float s = 0.f;
#pragma unroll
        for (int d = 0; d < DD; ++d) s += nes[d] * pbase[d * KTOT * UOUT + (kk0 + tl) * UOUT + o];
        Wp[o][tl] = s;                                    // transposed store
      }
      __syncthreads();
      for (int kk = 0; kk < 16; kk += 4) {
        int kl = kk0 + kk;
        v2f bop = ld2s(&Wp[cb + lm][kk + 2 * kh]);
#pragma unroll
        for (int rr = 0; rr < 2; ++rr) {
          int bRow = 16 * (rbase + rr) + lm;
          v2f aop = ld2s(&abase[bRow * astride + kl + 2 * kh]);
          acc[rr] = wmma4(aop, bop, acc[rr]);
        }
      }
    }
  }
#pragma unroll
  for (int rr = 0; rr < 2; ++rr) {
#pragma unroll
    for (int v = 0; v < 8; ++v) {
      int bRow = 16 * (rbase + rr) + v + 8 * kh;
      int col  = cb + lm;
      float hc = tanhf(acc[rr][v] + bias_s[col]);
      int bn = bRow * NN + n;
      float rg = rbuf[bn * HID + col];
      out[bn * HID + col] = rg * state[bn * HID + col] + (1.f - rg) * hc;
    }
  }
}

extern "C" void kernel_launch(void* const* d_in, const int* in_sizes, int n_in,
                              void* d_out, int out_size, void* d_ws, size_t ws_size,
                              hipStream_t stream) {
  (void)in_sizes; (void)n_in; (void)out_size; (void)ws_size;
  const float* x     = (const float*)d_in[0];
  const float* state = (const float*)d_in[1];
  const float* ne    = (const float*)d_in[2];
  const float* gw    = (const float*)d_in[3];
  const float* gb    = (const float*)d_in[4];
  const float* uw    = (const float*)d_in[5];
  const float* ub    = (const float*)d_in[6];
  float* out = (float*)d_out;
  float* ws  = (float*)d_ws;

  float* S1   = ws;                                 // 2048*2048
  float* S2   = S1  + (size_t)NN * NN;              // 2048*2048
  float* cat  = S2  + (size_t)NN * NN;              // 64*2048*96  ([b][n][96])
  float* xg1  = cat + (size_t)BB * NN * CIN;        // 64*2048*96  ([n][b][96])
  float* xg2  = xg1 + (size_t)BB * NN * CIN;        // 64*2048*96
  float* rbuf = xg2 + (size_t)BB * NN * CIN;        // 64*2048*64

  k_build_cat<<<(BB * NN * CIN) / 256, 256, 0, stream>>>(x, state, cat);
  k_supports <<<NN, 256, 0, stream>>>(ne, S1);
  k_cheb2    <<<dim3(NN / 128, NN / 64), 256, 0, stream>>>(S1, S2);
  // gate pass
  k_prop<<<dim3(NN / 128, BB), 256, 0, stream>>>(S1, cat, xg1);
  k_prop<<<dim3(NN / 128, BB), 256, 0, stream>>>(S2, cat, xg2);
  k_gate<<<NN, 256, 0, stream>>>(ne, gw, gb, cat, xg1, xg2, state, rbuf);
  // candidate pass (cat now holds [x, z*state])
  k_prop<<<dim3(NN / 128, BB), 256, 0, stream>>>(S1, cat, xg1);
  k_prop<<<dim3(NN / 128, BB), 256, 0, stream>>>(S2, cat, xg2);
  k_upd<<<NN, 256, 0, stream>>>(ne, uw, ub, cat, xg1, xg2, state, rbuf, out);
}